// CAWeightedFusion_2851858284711
// MI455X (gfx1250) — compile-verified
//
#include <hip/hip_runtime.h>
#include <hip/hip_bf16.h>

// ---------------------------------------------------------------------------
// CAWeightedFusion, collapsed to its exact closed form.
//
// reference output = sigmoid( relu( g @ W1^T ) @ W2^T ), where
//   g[b,c] = mean_{H,W}( ra + ca + rgb + chm )
// and ra/ca are softmax over the spatial axis N=4096 whose spatial mean is
// exactly 1/N each, independent of all attention inputs. Hence:
//   g[b,c] = 2/4096 + mean(rgb[b,c]) + mean(chm[b,c])
// The QKV GEMMs / cross-attention / softmaxes cancel out of the output.
// Cost floor = reading rgb+chm once: 134 MB -> ~5.8 us at 23.3 TB/s.
// Kernel 1 is that streaming reduction; kernel 2 runs the 8x512x24 MLP on
// V_WMMA_F32_16X16X4_F32 with a branch-free inner loop.
// ---------------------------------------------------------------------------

typedef __attribute__((ext_vector_type(2))) float v2f;
typedef __attribute__((ext_vector_type(8))) float v8f;

#define N_SPATIAL 4096   // H*W
#define C_CH      512
#define B_BATCH   8
#define HID_DIM   24

// ---------------------------------------------------------------------------
// Kernel 1: per-(b,c) spatial mean of rgb+chm, plus the 2/N softmax constant.
// One block per channel-row; 256 threads stream 2 x 16KB with b128 loads.
// Reduction: wave32 shfl_xor butterfly, then one LDS pass over 8 waves.
// ---------------------------------------------------------------------------
__global__ __launch_bounds__(256) void ca_channel_mean_kernel(
    const float* __restrict__ rgb,
    const float* __restrict__ chm,
    float* __restrict__ g)                    // [B*C] = [4096]
{
    const int bc  = blockIdx.x;               // b*512 + c
    const int tid = threadIdx.x;              // 0..255

    const float4* r4 = (const float4*)(rgb + (size_t)bc * N_SPATIAL);
    const float4* c4 = (const float4*)(chm + (size_t)bc * N_SPATIAL);

    float s = 0.0f;
    #pragma unroll
    for (int i = 0; i < N_SPATIAL / (256 * 4); ++i) {   // 4 iterations
        float4 a = r4[tid + 256 * i];
        float4 b = c4[tid + 256 * i];
        s += (a.x + a.y) + (a.z + a.w) + (b.x + b.y) + (b.z + b.w);
    }

    // wave32 butterfly reduction (no barriers)
    #pragma unroll
    for (int off = 16; off > 0; off >>= 1)
        s += __shfl_xor(s, off, 32);

    __shared__ float wsum[8];
    if ((tid & 31) == 0) wsum[tid >> 5] = s;
    __syncthreads();
    if (tid == 0) {
        float t = ((wsum[0] + wsum[1]) + (wsum[2] + wsum[3])) +
                  ((wsum[4] + wsum[5]) + (wsum[6] + wsum[7]));
        g[bc] = t * (1.0f / N_SPATIAL) + 2.0f / N_SPATIAL;
    }
}

// ---------------------------------------------------------------------------
// Kernel 2: gate = sigmoid( relu(g @ W1^T) @ W2^T ) via f32 WMMA.
// Single wave32. H1[16x32] tile = A[16x4] x B[4x16] over K=512 (2 N-tiles).
//
// VGPR layouts (CDNA5 ISA 7.12.2):
//   A 16x4 f32 (2 VGPRs): lane&15 = M; lanes>=16 hold K+2,K+3.
//   B 4x16  f32 (2 VGPRs): lane&15 = N; lanes>=16 hold rows K+2,K+3.
//   C/D 16x16 f32 (8 VGPRs): VGPR v -> M = v + 8*(lane>=16), N = lane&15.
//
// Branch-free trick: D row m depends only on A row m, D col n only on B col n.
// The downstream dot reads only b<8, j<24, so A rows 8..15 and B1 cols 8..15
// may hold arbitrary (but memory-safe) data — we clamp the row index instead
// of predicating the loads. Inner loop = 3x global_load_b64 + 2x v_wmma.
// EXEC is all-1s at every WMMA (uniform control flow, one full wave).
// ---------------------------------------------------------------------------
__global__ __launch_bounds__(32) void ca_gate_mlp_wmma_kernel(
    const float* __restrict__ g,      // [8,512]
    const float* __restrict__ w1,     // [24,512] row-major
    const float* __restrict__ w2,     // [1,24]
    float* __restrict__ out)          // [8]
{
    const int lane = threadIdx.x;          // 0..31
    const int row  = lane & 15;            // M for A, N for B/D
    const int koff = (lane >> 4) * 2;      // 0 (low half) or 2 (high half)

    // Clamped, always-valid base pointers (see note above).
    const float* gp  = g  + (row & 7) * C_CH + koff;          // A row (clamped)
    const float* b0p = w1 + row * C_CH + koff;                // B0 col: j = 0..15
    const float* b1p = w1 + (16 + (row & 7)) * C_CH + koff;   // B1 col: j = 16..23

    v8f c0 = {};   // N-tile 0: hidden units j = 0..15
    v8f c1 = {};   // N-tile 1: hidden units j = 16..23 (cols 8..15 unused)

    #pragma unroll 8
    for (int k = 0; k < C_CH; k += 4) {
        v2f a  = *(const v2f*)(gp  + k);
        v2f b0 = *(const v2f*)(b0p + k);
        v2f b1 = *(const v2f*)(b1p + k);
        // (neg_a, A, neg_b, B, c_mod, C, reuse_a, reuse_b)
        c0 = __builtin_amdgcn_wmma_f32_16x16x4_f32(
                false, a, false, b0, (short)0, c0, false, false);
        c1 = __builtin_amdgcn_wmma_f32_16x16x4_f32(
                false, a, false, b1, (short)0, c1, false, false);
    }

    // Scatter D tiles (with ReLU) to LDS: h1s[m][j], m=batch, j=hidden unit.
    __shared__ float h1s[16][32];
    const int mbase = (lane >> 4) * 8;
    #pragma unroll
    for (int v = 0; v < 8; ++v) {
        const int m = mbase + v;
        h1s[m][row]      = fmaxf(c0[v], 0.0f);
        h1s[m][16 + row] = fmaxf(c1[v], 0.0f);
    }
    __syncthreads();

    if (lane < B_BATCH) {
        float acc = 0.0f;
        #pragma unroll
        for (int j = 0; j < HID_DIM; ++j)
            acc += w2[j] * h1s[lane][j];
        out[lane] = 1.0f / (1.0f + __expf(-acc));
    }
}

// ---------------------------------------------------------------------------
// Host entry.
// Inputs: 0=rgb 1=chm 2=w_rgb_qkv 3=b_rgb_qkv 4=w_chm_qkv 5=b_chm_qkv
//         6=w_mlp1 7=w_mlp2   (2..5 provably do not affect the output)
// ---------------------------------------------------------------------------
extern "C" void kernel_launch(void* const* d_in, const int* in_sizes, int n_in,
                              void* d_out, int out_size, void* d_ws, size_t ws_size,
                              hipStream_t stream) {
    const float* rgb = (const float*)d_in[0];
    const float* chm = (const float*)d_in[1];
    const float* w1  = (const float*)d_in[6];
    const float* w2  = (const float*)d_in[7];

    float* g   = (float*)d_ws;     // 4096 floats = 16 KB scratch
    float* out = (float*)d_out;    // 8 floats

    ca_channel_mean_kernel<<<B_BATCH * C_CH, 256, 0, stream>>>(rgb, chm, g);
    ca_gate_mlp_wmma_kernel<<<1, 32, 0, stream>>>(g, w1, w2, out);
}